// SumFlattenedOuterProduct_70016556860029
// MI455X (gfx1250) — compile-verified
//
#include <hip/hip_runtime.h>

typedef float v2f __attribute__((ext_vector_type(2)));
typedef float v4f __attribute__((ext_vector_type(4)));
typedef float v8f __attribute__((ext_vector_type(8)));

#define S_DIM 2048
#define B_DIM 16
#define D1 256
#define D2 256
#define ROWSTRIDE (B_DIM * D1)      // 4096 floats per s step

#define BM 128
#define BN 64
#define BK 32
#define NKT (S_DIM / BK)            // 64 K-tiles
#define LDSK 36                     // padded k-stride (dwords): b64 frags conflict-free

__global__ __launch_bounds__(256)
void SumFlattenedOuterProduct_kernel(const float* __restrict__ gA,  // [S,B,D1]
                                     const float* __restrict__ gB,  // [S,B,D2]
                                     float* __restrict__ gC)        // [B, D1*D2]
{
    // double-buffered, transposed [m][k] tiles
    __shared__ float lA[2][BM * LDSK];
    __shared__ float lB[2][BN * LDSK];

    const int b  = blockIdx.z;
    const int n0 = blockIdx.x * BN;
    const int m0 = blockIdx.y * BM;

    const int t    = threadIdx.x;
    const int lane = t & 31;
    const int wave = t >> 5;
    const int wmo  = (wave >> 1) * 32;   // 4 waves along m
    const int wno  = (wave & 1) * 32;    // 2 waves along n

    // WMMA f32 16x16x4 A/B fragment lane map: m(or n) = lane&15, k base = (lane>>4)*2
    const int fl_mn = lane & 15;
    const int fl_k  = (lane >> 4) * 2;

    const float* pA = gA + (size_t)b * D1 + m0;
    const float* pB = gB + (size_t)b * D2 + n0;

    // staging coordinates (constant per thread): each thread owns 4 A-slots and
    // 2 B-slots; a slot is (m row, k-quad) -> 4 consecutive k values.
    int am[4], akq[4];
#pragma unroll
    for (int q = 0; q < 4; ++q) { int s = t + q * 256; am[q] = s & 127; akq[q] = s >> 7; }
    int bm[2], bkq[2];
#pragma unroll
    for (int q = 0; q < 2; ++q) { int s = t + q * 256; bm[q] = s & 63; bkq[q] = s >> 6; }

    float ra[16];
    float rb[8];

    auto loadG = [&](int s0) {
#pragma unroll
        for (int q = 0; q < 4; ++q)
#pragma unroll
            for (int j = 0; j < 4; ++j)
                ra[q * 4 + j] = pA[(size_t)(s0 + akq[q] * 4 + j) * ROWSTRIDE + am[q]];
#pragma unroll
        for (int q = 0; q < 2; ++q)
#pragma unroll
            for (int j = 0; j < 4; ++j)
                rb[q * 4 + j] = pB[(size_t)(s0 + bkq[q] * 4 + j) * ROWSTRIDE + bm[q]];
        // warm L2 one tile further ahead (global_prefetch_b8)
        if (s0 + BK < S_DIM) {
            __builtin_prefetch(&pA[(size_t)(s0 + BK + akq[0] * 4) * ROWSTRIDE + am[0]], 0, 1);
            __builtin_prefetch(&pB[(size_t)(s0 + BK + bkq[0] * 4) * ROWSTRIDE + bm[0]], 0, 1);
        }
    };

    auto storeS = [&](int buf) {
#pragma unroll
        for (int q = 0; q < 4; ++q) {
            v4f v = { ra[q * 4 + 0], ra[q * 4 + 1], ra[q * 4 + 2], ra[q * 4 + 3] };
            *(v4f*)&lA[buf][am[q] * LDSK + akq[q] * 4] = v;
        }
#pragma unroll
        for (int q = 0; q < 2; ++q) {
            v4f v = { rb[q * 4 + 0], rb[q * 4 + 1], rb[q * 4 + 2], rb[q * 4 + 3] };
            *(v4f*)&lB[buf][bm[q] * LDSK + bkq[q] * 4] = v;
        }
    };

    v8f acc[2][2];
#pragma unroll
    for (int i = 0; i < 2; ++i)
#pragma unroll
        for (int j = 0; j < 2; ++j)
            acc[i][j] = v8f{};

    auto compute = [&](int buf) {
#pragma unroll
        for (int kk = 0; kk < BK; kk += 4) {
            v2f af[2], bf[2];
#pragma unroll
            for (int mt = 0; mt < 2; ++mt)
                af[mt] = *(const v2f*)&lA[buf][(wmo + mt * 16 + fl_mn) * LDSK + kk + fl_k];
#pragma unroll
            for (int nt = 0; nt < 2; ++nt)
                bf[nt] = *(const v2f*)&lB[buf][(wno + nt * 16 + fl_mn) * LDSK + kk + fl_k];
#pragma unroll
            for (int mt = 0; mt < 2; ++mt)
#pragma unroll
                for (int nt = 0; nt < 2; ++nt)
                    acc[mt][nt] = __builtin_amdgcn_wmma_f32_16x16x4_f32(
                        false, af[mt], false, bf[nt],
                        (short)0, acc[mt][nt], false, false);
        }
    };

    // ---- software pipeline: global loads for tile k+1 overlap compute on tile k ----
    loadG(0);
    storeS(0);
    __syncthreads();

    for (int kt = 0; kt < NKT; ++kt) {
        const int cur = kt & 1;
        if (kt + 1 < NKT) loadG((kt + 1) * BK);   // issue early; waits land after WMMAs
        compute(cur);
        if (kt + 1 < NKT) {
            storeS(cur ^ 1);
            __syncthreads();
        }
    }

    // ---- store: C/D layout: VGPR r, lane L -> M = r + (L>>4)*8, N = L&15 ----
    float* pC = gC + (size_t)b * (D1 * D2);
    const int mhalf = (lane >> 4) * 8;
#pragma unroll
    for (int mt = 0; mt < 2; ++mt)
#pragma unroll
        for (int nt = 0; nt < 2; ++nt)
#pragma unroll
            for (int r = 0; r < 8; ++r) {
                int m = m0 + wmo + mt * 16 + r + mhalf;
                int n = n0 + wno + nt * 16 + fl_mn;
                pC[(size_t)m * D2 + n] = acc[mt][nt][r];
            }
}

extern "C" void kernel_launch(void* const* d_in, const int* in_sizes, int n_in,
                              void* d_out, int out_size, void* d_ws, size_t ws_size,
                              hipStream_t stream) {
    const float* in1 = (const float*)d_in[0];   // [2048,16,256] f32
    const float* in2 = (const float*)d_in[1];   // [2048,16,256] f32
    float* out = (float*)d_out;                 // [16, 65536] f32

    dim3 grid(D2 / BN, D1 / BM, B_DIM);         // (4, 2, 16) = 128 workgroups
    dim3 block(256);
    SumFlattenedOuterProduct_kernel<<<grid, block, 0, stream>>>(in1, in2, out);
}